// GF_train_33071248179252
// MI455X (gfx1250) — compile-verified
//
#include <hip/hip_runtime.h>

typedef __attribute__((ext_vector_type(2))) float v2f;
typedef __attribute__((ext_vector_type(4))) float v4f;
typedef __attribute__((ext_vector_type(8))) float v8f;

#define GNUM 60      // rotation group size
#define KNEI 13      // neighbors per group element
#define KCHUNK 52    // 4 channels * 13 taps staged per LDS slab
#define BATCH 256

// ---------------------------------------------------------------------------
// Fused comb_conv (+bias, +BN/ReLU, +residual) using V_WMMA_F32_16X16X4_F32.
// Block: 256 threads = 8 wave32s; computes a [64 x NT] tile of out[:, :, g].
// Waves: 4 (M) x 2 (N), each wave holds NACC 16x16 f32 accumulators.
// A is gathered (via nei) into LDS row-major [64][52]; B is staged transposed
// [NT][52] directly from W (W[o,c,k] flat == [O][C*13]) with b128 loads, so
// both WMMA fragments are single ds_load_b64 per lane.
// ---------------------------------------------------------------------------
template <int CIN, int OTOT, int NACC, bool BNRELU, bool RESID>
__global__ __launch_bounds__(256)
void comb_conv_wmma(const float* __restrict__ X,     // [B, CIN, 60]
                    const float* __restrict__ W,     // [OTOT, CIN, 13]
                    const float* __restrict__ bias,  // [OTOT]
                    const float* __restrict__ gamma, // [OTOT] (BNRELU)
                    const float* __restrict__ beta,  // [OTOT] (BNRELU)
                    const float* __restrict__ resid, // [B, OTOT, 60] (RESID)
                    float* __restrict__ Y,           // [B, OTOT, 60]
                    const int* __restrict__ nei)     // [60*13]
{
    constexpr int NT    = NACC * 32;   // block N tile (two wave columns)
    constexpr int CKTOT = CIN * KNEI;  // full contraction length

    __shared__ __align__(16) float Asm[64 * KCHUNK];  // [m][q]
    __shared__ __align__(16) float Bsm[NT * KCHUNK];  // [n][q]  (transposed)
    __shared__ int snei[KNEI];

    const int g  = blockIdx.x;
    const int o0 = blockIdx.y * NT;
    const int b0 = blockIdx.z * 64;

    const int tid   = threadIdx.x;
    const int wave  = tid >> 5;
    const int lane  = tid & 31;
    const int lhalf = lane >> 4;   // 0|1 : half-wave (K offset / M offset)
    const int l16   = lane & 15;

    const int mwave = wave & 3;    // 0..3 -> 16-row tiles
    const int nwave = wave >> 2;   // 0..1 -> NACC*16-col strips

    if (tid < KNEI) snei[tid] = nei[g * KNEI + tid];
    __syncthreads();

    v8f acc[NACC];
#pragma unroll
    for (int j = 0; j < NACC; ++j) acc[j] = (v8f){0,0,0,0,0,0,0,0};

    for (int c0 = 0; c0 < CIN; c0 += 4) {
        // Stage gathered A slab: A[m][cl*13+k] = X[b0+m, c0+cl, nei[g,k]]
        for (int idx = tid; idx < 64 * KCHUNK; idx += 256) {
            int m  = idx / KCHUNK;
            int q  = idx - m * KCHUNK;
            int cl = q / KNEI;
            int k  = q - cl * KNEI;
            Asm[idx] = X[((size_t)(b0 + m) * CIN + (c0 + cl)) * GNUM + snei[k]];
        }
        // Stage B slab transposed with aligned 128b loads:
        //   Bsm[n][qq..qq+3] = W[o0+n, c0*13 + qq .. +3]
        for (int idx = tid; idx < NT * (KCHUNK / 4); idx += 256) {
            int n  = idx / (KCHUNK / 4);
            int qq = (idx - n * (KCHUNK / 4)) * 4;
            const v4f wv = *(const v4f*)(&W[(size_t)(o0 + n) * CKTOT + c0 * KNEI + qq]);
            *(v4f*)(&Bsm[n * KCHUNK + qq]) = wv;
        }
        __syncthreads();

        const float* arow = &Asm[(mwave * 16 + l16) * KCHUNK];
#pragma unroll
        for (int kk = 0; kk < KCHUNK; kk += 4) {
            const int kq = kk + lhalf * 2;                 // fragment K offset
            const v2f a = *(const v2f*)(arow + kq);        // ds_load_b64
#pragma unroll
            for (int j = 0; j < NACC; ++j) {
                const int n = nwave * (NACC * 16) + j * 16 + l16;
                const v2f b = *(const v2f*)(&Bsm[n * KCHUNK + kq]);  // ds_load_b64
                acc[j] = __builtin_amdgcn_wmma_f32_16x16x4_f32(
                    false, a, false, b, (short)0, acc[j], false, false);
            }
        }
        __syncthreads();
    }

    // Epilogue: bias (+BN/ReLU) (+residual). C/D layout: VGPR i -> M=i+8*lhalf.
#pragma unroll
    for (int j = 0; j < NACC; ++j) {
        const int o = o0 + nwave * (NACC * 16) + j * 16 + l16;
        const float bi = bias[o];
        float ga = 0.f, bt = 0.f;
        if constexpr (BNRELU) { ga = gamma[o]; bt = beta[o]; }
#pragma unroll
        for (int i = 0; i < 8; ++i) {
            const int m = b0 + mwave * 16 + lhalf * 8 + i;
            float v = acc[j][i] + bi;
            if constexpr (BNRELU) {
                v = v * ga + bt;
                v = v > 0.f ? v : 0.f;
            }
            const size_t oidx = ((size_t)m * OTOT + o) * GNUM + g;
            if constexpr (RESID) v += resid[oidx];
            Y[oidx] = v;
        }
    }
}

// ---------------------------------------------------------------------------
// Per-sample finalize: inv = mean_g(eqv_raw); normalize eqv per (b,g) over f,
// normalize inv per b over f. One block (64 threads) per b.
// ---------------------------------------------------------------------------
__global__ __launch_bounds__(64)
void finalize_eqv_kernel(const float* __restrict__ eraw,  // [B,32,60]
                         float* __restrict__ eqv_out,     // [B,32,60]
                         float* __restrict__ inv_out)     // [B,32]
{
    __shared__ float sm[32 * GNUM];
    __shared__ float rng[GNUM];
    __shared__ float sinv[32];
    __shared__ float rni;

    const int b   = blockIdx.x;
    const int tid = threadIdx.x;

    for (int i = tid; i < 32 * GNUM; i += 64)
        sm[i] = eraw[(size_t)b * 32 * GNUM + i];
    __syncthreads();

    if (tid < GNUM) {
        float s = 0.f;
#pragma unroll
        for (int f = 0; f < 32; ++f) { float v = sm[f * GNUM + tid]; s += v * v; }
        rng[tid] = 1.0f / fmaxf(sqrtf(s), 1e-4f);
    }
    __syncthreads();
    if (tid < 32) {
        float s = 0.f;
        for (int gg = 0; gg < GNUM; ++gg) s += sm[tid * GNUM + gg];
        sinv[tid] = s * (1.0f / GNUM);
    }
    __syncthreads();
    if (tid == 0) {
        float s = 0.f;
#pragma unroll
        for (int f = 0; f < 32; ++f) s += sinv[f] * sinv[f];
        rni = 1.0f / fmaxf(sqrtf(s), 1e-4f);
    }
    __syncthreads();

    for (int i = tid; i < 32 * GNUM; i += 64)
        eqv_out[(size_t)b * 32 * GNUM + i] = sm[i] * rng[i % GNUM];
    if (tid < 32)
        inv_out[(size_t)b * 32 + tid] = sinv[tid] * rni;
}

// ---------------------------------------------------------------------------
// des2dr: cor[b,a] = sum_{f,g} e0[b,f,permu[a*60+g]] * e1[b,f,g]; argmax_a.
// One block (64 threads) per b.
// ---------------------------------------------------------------------------
__global__ __launch_bounds__(64)
void des2dr_kernel(const float* __restrict__ e0, const float* __restrict__ e1,
                   const int* __restrict__ permu,
                   float* __restrict__ pre_f, int* __restrict__ pre_i)
{
    __shared__ float s0[32 * GNUM];
    __shared__ float s1[32 * GNUM];
    __shared__ float cor[GNUM];

    const int b   = blockIdx.x;
    const int tid = threadIdx.x;

    for (int i = tid; i < 32 * GNUM; i += 64) {
        s0[i] = e0[(size_t)b * 32 * GNUM + i];
        s1[i] = e1[(size_t)b * 32 * GNUM + i];
    }
    __syncthreads();

    if (tid < GNUM) {
        float s = 0.f;
        for (int gg = 0; gg < GNUM; ++gg) {
            const int p = permu[tid * GNUM + gg];
#pragma unroll
            for (int f = 0; f < 32; ++f)
                s += s0[f * GNUM + p] * s1[f * GNUM + gg];
        }
        cor[tid] = s;
    }
    __syncthreads();
    if (tid == 0) {
        int best = 0;
        float bv = cor[0];
        for (int a = 1; a < GNUM; ++a)
            if (cor[a] > bv) { bv = cor[a]; best = a; }  // first-max, like argmax
        pre_f[b] = (float)best;
        pre_i[b] = best;
    }
}

// ---------------------------------------------------------------------------
// ability = mean(pre == true); also emit true_idx as float. One block, B=256.
// ---------------------------------------------------------------------------
__global__ __launch_bounds__(256)
void ability_kernel(const int* __restrict__ pre, const int* __restrict__ tru,
                    float* __restrict__ ability, float* __restrict__ true_f)
{
    __shared__ int cnt[256];
    const int tid = threadIdx.x;
    true_f[tid] = (float)tru[tid];
    cnt[tid] = (pre[tid] == tru[tid]) ? 1 : 0;
    __syncthreads();
    for (int s = 128; s > 0; s >>= 1) {
        if (tid < s) cnt[tid] += cnt[tid + s];
        __syncthreads();
    }
    if (tid == 0) *ability = (float)cnt[0] * (1.0f / 256.0f);
}

// ---------------------------------------------------------------------------
extern "C" void kernel_launch(void* const* d_in, const int* in_sizes, int n_in,
                              void* d_out, int out_size, void* d_ws, size_t ws_size,
                              hipStream_t stream)
{
    (void)in_sizes; (void)n_in; (void)out_size; (void)ws_size;

    const float* feats0   = (const float*)d_in[0];
    const float* feats1   = (const float*)d_in[1];
    const int*   true_idx = (const int*)d_in[2];
    const int*   nei      = (const int*)d_in[3];
    const int*   permu    = (const int*)d_in[4];
    const float* W_in = (const float*)d_in[5];  const float* b_in = (const float*)d_in[6];
    const float* W1   = (const float*)d_in[7];  const float* b1   = (const float*)d_in[8];
    const float* g1   = (const float*)d_in[9];  const float* be1  = (const float*)d_in[10];
    const float* W2   = (const float*)d_in[11]; const float* b2   = (const float*)d_in[12];
    const float* g2   = (const float*)d_in[13]; const float* be2  = (const float*)d_in[14];
    const float* Wo   = (const float*)d_in[15]; const float* bo   = (const float*)d_in[16];
    const float* go   = (const float*)d_in[17]; const float* beo  = (const float*)d_in[18];

    // Workspace layout (floats)
    float* ws = (float*)d_ws;
    size_t p = 0;
    float* dbuf  = ws + p; p += (size_t)BATCH * 256 * GNUM; // 3932160
    float* fcbuf = ws + p; p += (size_t)BATCH * 512 * GNUM; // 7864320
    float* d2buf = ws + p; p += (size_t)BATCH * 256 * GNUM; // 3932160
    float* eraw  = ws + p; p += (size_t)BATCH * 32  * GNUM; // 491520
    int*   prei  = (int*)(ws + p);

    // Output layout (floats), in tuple return order
    float* out  = (float*)d_out;
    const size_t F = (size_t)BATCH * 32 * GNUM;   // 491520
    float* o_f0 = out;
    float* o_f1 = out + F;
    float* o_e0 = out + 2 * F;
    float* o_e1 = out + 3 * F;
    float* o_i0 = out + 4 * F;
    float* o_i1 = o_i0 + (size_t)BATCH * 32;
    float* o_ab = o_i1 + (size_t)BATCH * 32;
    float* o_tr = o_ab + 1;
    float* o_pr = o_tr + BATCH;

    // feats passthrough (graph-capture-safe async D2D)
    hipMemcpyAsync(o_f0, feats0, F * sizeof(float), hipMemcpyDeviceToDevice, stream);
    hipMemcpyAsync(o_f1, feats1, F * sizeof(float), hipMemcpyDeviceToDevice, stream);

    auto run_net = [&](const float* feats, float* eqv_out, float* inv_out) {
        dim3 blk(256);
        // conv_in: 32 -> 256 (bias only)
        comb_conv_wmma<32, 256, 2, false, false>
            <<<dim3(GNUM, 4, 4), blk, 0, stream>>>(feats, W_in, b_in,
                nullptr, nullptr, nullptr, dbuf, nei);
        // conv1: 256 -> 512, BN+ReLU
        comb_conv_wmma<256, 512, 2, true, false>
            <<<dim3(GNUM, 8, 4), blk, 0, stream>>>(dbuf, W1, b1,
                g1, be1, nullptr, fcbuf, nei);
        // conv2: 512 -> 256, BN+ReLU, + residual d
        comb_conv_wmma<512, 256, 2, true, true>
            <<<dim3(GNUM, 4, 4), blk, 0, stream>>>(fcbuf, W2, b2,
                g2, be2, dbuf, d2buf, nei);
        // conv_out: 256 -> 32, BN+ReLU, + feats
        comb_conv_wmma<256, 32, 1, true, true>
            <<<dim3(GNUM, 1, 4), blk, 0, stream>>>(d2buf, Wo, bo,
                go, beo, feats, eraw, nei);
        finalize_eqv_kernel<<<BATCH, 64, 0, stream>>>(eraw, eqv_out, inv_out);
    };

    run_net(feats0, o_e0, o_i0);
    run_net(feats1, o_e1, o_i1);

    des2dr_kernel<<<BATCH, 64, 0, stream>>>(o_e0, o_e1, permu, o_pr, prei);
    ability_kernel<<<1, 256, 0, stream>>>(prei, true_idx, o_ab, o_tr);
}